// SigmoidalTransformer_51934744543566
// MI455X (gfx1250) — compile-verified
//
#include <hip/hip_runtime.h>
#include <math.h>

// Problem constants (match reference)
#define ND   16
#define NLAY 8
#define BQ   8
#define TQ   4096
#define DQ   1024
#define OUTQ 384                 // 3*ND*NLAY
#define BT   (BQ * TQ)           // 32768

// Tiling
#define ROWS_PER_BLOCK 64
#define KCH 32
#define NCHUNK (DQ / KCH)        // 32
#define LDA_STRIDE 40            // 32 k + pad, 160B row stride (16B aligned)
#define LDB_STRIDE 40
#define P_STRIDE   388           // 384 + 4 pad (bank spread)
#define A_ELEMS (ROWS_PER_BLOCK * LDA_STRIDE)   // 2560 floats
#define B_ELEMS (OUTQ * LDB_STRIDE)             // 15360 floats
#define NBLOCKS (BT / ROWS_PER_BLOCK)           // 512

typedef float v2f __attribute__((ext_vector_type(2)));
typedef float v4f __attribute__((ext_vector_type(4)));
typedef float v8f __attribute__((ext_vector_type(8)));

#define AS1 __attribute__((address_space(1)))
#define AS3 __attribute__((address_space(3)))

typedef AS3 float lfloat;        // LDS-resident float
typedef AS3 v2f   lv2f;
typedef AS3 v4f   lv4f;

typedef int v4i_t __attribute__((vector_size(16)));
typedef AS1 v4i_t* gptr128_t;    // global-AS pointer to 16B vector
typedef AS3 v4i_t* lptr128_t;    // LDS-AS pointer to 16B vector

#if __has_builtin(__builtin_amdgcn_global_load_async_to_lds_b128) && \
    __has_builtin(__builtin_amdgcn_s_wait_asynccnt)
#define HAVE_ASYNC_LDS 1
#else
#define HAVE_ASYNC_LDS 0
#endif

// 16-byte copy global -> LDS; async (ASYNCcnt-tracked, no VGPR bounce) when available.
__device__ __forceinline__ void copy16_g2l(const float* gp, lfloat* lp) {
#if HAVE_ASYNC_LDS
  __builtin_amdgcn_global_load_async_to_lds_b128(
      (gptr128_t)(uintptr_t)gp,          // global src (inttoptr: flat==global)
      (lptr128_t)lp,                     // LDS dst (bitcast within AS3)
      0, 0);
#else
  *(lv4f*)lp = *(const v4f*)gp;
#endif
}

__device__ __forceinline__ void wait_g2l() {
#if HAVE_ASYNC_LDS
  __builtin_amdgcn_s_wait_asynccnt(0);
#endif
}

__device__ __forceinline__ float sp_f(float v) {          // softplus, stable
  return fmaxf(v, 0.0f) + log1pf(expf(-fabsf(v)));
}
__device__ __forceinline__ float logsig_f(float v) {      // log sigmoid, stable
  return fminf(v, 0.0f) - log1pf(expf(-fabsf(v)));
}

__global__ __launch_bounds__(256, 1)
void sigflow_fused_kernel(const float* __restrict__ x,
                          const float* __restrict__ cond,
                          const float* __restrict__ mask,
                          const float* __restrict__ W,
                          const float* __restrict__ bias,
                          float* __restrict__ out,
                          float* __restrict__ partials)
{
  extern __shared__ float smem_g[];
  lfloat* smemL = (lfloat*)smem_g;                  // one addrspacecast; AS3 after this
  lfloat* lA0 = smemL;                              // 2 x A buffers
  lfloat* lB0 = smemL + 2 * A_ELEMS;                // 2 x B buffers
  lfloat* lP  = smemL + 2 * A_ELEMS + 2 * B_ELEMS;  // 64 * 388 params

  const int tid = threadIdx.x;
  const int rowBase = blockIdx.x * ROWS_PER_BLOCK;

  const int lane = tid & 31;
  const int wv   = tid >> 5;     // 0..7
  const int half = lane >> 4;    // 0/1
  const int r16  = lane & 15;
  const int mt   = wv & 3;       // M tile (16 rows each)
  const int nh   = wv >> 2;      // N half (12 tiles each)

  // Per-thread staging slots (precomputed)
  const int arow0 = tid >> 3;                 // A/B slots: +32 per iteration
  const int ako   = (tid & 7) << 2;

  v8f acc[12];
  #pragma unroll
  for (int j = 0; j < 12; ++j)
    #pragma unroll
    for (int i = 0; i < 8; ++i) acc[j][i] = 0.0f;

  // ---- prologue: issue chunk 0 into buffer 0 ----
  {
    #pragma unroll
    for (int it = 0; it < 2; ++it) {
      int row = arow0 + it * 32;
      copy16_g2l(cond + (size_t)(rowBase + row) * DQ + ako,
                 lA0 + row * LDA_STRIDE + ako);
    }
    #pragma unroll
    for (int it = 0; it < 12; ++it) {
      int col = arow0 + it * 32;
      copy16_g2l(W + (size_t)col * DQ + ako,
                 lB0 + col * LDB_STRIDE + ako);
    }
  }

  for (int c = 0; c < NCHUNK; ++c) {
    const int buf = c & 1;
    lfloat* curA = lA0 + buf * A_ELEMS;
    lfloat* curB = lB0 + buf * B_ELEMS;
    lfloat* nxtA = lA0 + (buf ^ 1) * A_ELEMS;
    lfloat* nxtB = lB0 + (buf ^ 1) * B_ELEMS;

    wait_g2l();            // my async fills of buf complete (ASYNCcnt==0)
    __syncthreads();       // all waves' fills visible; prev compute on buf^1 done

    if (c + 1 < NCHUNK) {  // issue next chunk into alternate buffer
      const int kb = (c + 1) * KCH;
      #pragma unroll
      for (int it = 0; it < 2; ++it) {
        int row = arow0 + it * 32;
        copy16_g2l(cond + (size_t)(rowBase + row) * DQ + kb + ako,
                   nxtA + row * LDA_STRIDE + ako);
      }
      #pragma unroll
      for (int it = 0; it < 12; ++it) {
        int col = arow0 + it * 32;
        copy16_g2l(W + (size_t)col * DQ + kb + ako,
                   nxtB + col * LDB_STRIDE + ako);
      }
    }

    // ---- WMMA f32 16x16x4: 8 k-steps, A fragment reused across 12 N tiles
    const lfloat* pa  = curA + (mt * 16 + r16) * LDA_STRIDE + 2 * half;
    const lfloat* pb0 = curB + (nh * 192 + r16) * LDB_STRIDE + 2 * half;
    #pragma unroll
    for (int kk = 0; kk < 8; ++kk) {
      v2f a = *(const lv2f*)(pa + 4 * kk);
      #pragma unroll
      for (int j = 0; j < 12; ++j) {
        v2f b = *(const lv2f*)(pb0 + j * 16 * LDB_STRIDE + 4 * kk);
        acc[j] = __builtin_amdgcn_wmma_f32_16x16x4_f32(
            false, a, false, b, (short)0, acc[j], false, false);
      }
    }
  }
  __syncthreads();

  // ---- epilogue: params(+bias) -> LDS [row][col] ----
  #pragma unroll
  for (int j = 0; j < 12; ++j) {
    int col = (nh * 12 + j) * 16 + r16;
    float bv = bias[col];
    #pragma unroll
    for (int i = 0; i < 8; ++i) {
      int m = mt * 16 + i + 8 * half;
      lP[m * P_STRIDE + col] = acc[j][i] + bv;
    }
  }
  __syncthreads();

  // ---- sigmoidal-flow chain, one thread per row ----
  float ldpart = 0.0f;
  if (tid < ROWS_PER_BLOCK) {
    int row = rowBase + tid;
    float mk = mask[row];
    float z  = x[row] * mk;
    const lfloat* pr = lP + tid * P_STRIDE;
    #pragma unroll 1
    for (int l = 0; l < NLAY; ++l) {
      const lfloat* p = pr + l * 48;
      float wl[ND], ew[ND];
      float wmax = -3.4e38f;
      #pragma unroll
      for (int j = 0; j < ND; ++j) { wl[j] = p[32 + j] * mk; wmax = fmaxf(wmax, wl[j]); }
      float wsum = 0.0f;
      #pragma unroll
      for (int j = 0; j < ND; ++j) { ew[j] = expf(wl[j] - wmax); wsum += ew[j]; }
      float logwsum  = logf(wsum);
      float inv_wsum = 1.0f / wsum;

      float xpre = 0.0f;
      float lj[ND]; float lmax = -3.4e38f;
      #pragma unroll
      for (int j = 0; j < ND; ++j) {
        float a   = sp_f(p[j] * mk);
        float pre = a * z + p[ND + j] * mk;
        float sg  = 1.0f / (1.0f + expf(-pre));
        xpre += (ew[j] * inv_wsum) * sg;
        float v = (wl[j] - wmax - logwsum) + logsig_f(pre) + logsig_f(-pre) + logf(a);
        lj[j] = v; lmax = fmaxf(lmax, v);
      }
      float s = 0.0f;
      #pragma unroll
      for (int j = 0; j < ND; ++j) s += expf(lj[j] - lmax);
      float lse = lmax + logf(s);

      float xpc = xpre * (1.0f - 1e-6f) + 0.5e-6f;
      float lx  = logf(xpc);
      float l1  = log1pf(-xpc);
      z = lx - l1;
      float lde = lse + (-1.0000005e-06f) - (lx + l1);   // + log(1-DELTA)
      ldpart += lde * mk;
    }
    out[row]      = z;
    out[BT + row] = mk;
  }

  // ---- deterministic block reduction of logdet partials ----
  __syncthreads();
  lfloat* red = lA0;                       // reuse staging LDS
  if (tid < ROWS_PER_BLOCK) red[tid] = ldpart;
  __syncthreads();
  if (tid == 0) {
    float s = 0.0f;
    for (int i = 0; i < ROWS_PER_BLOCK; ++i) s += red[i];
    partials[blockIdx.x] = s;
  }
}

__global__ void sigflow_reduce_logdet(const float* __restrict__ partials,
                                      const float* __restrict__ logdet_in,
                                      float* __restrict__ out)
{
  int b = threadIdx.x;
  if (b < BQ) {
    float s = logdet_in[b];
    const int per_batch = NBLOCKS / BQ;    // 64 blocks per batch
    for (int i = 0; i < per_batch; ++i) s += partials[b * per_batch + i];
    out[2 * BT + b] = s;
  }
}

extern "C" void kernel_launch(void* const* d_in, const int* in_sizes, int n_in,
                              void* d_out, int out_size, void* d_ws, size_t ws_size,
                              hipStream_t stream)
{
  (void)in_sizes; (void)n_in; (void)out_size; (void)ws_size;
  const float* x      = (const float*)d_in[0];
  const float* cond   = (const float*)d_in[1];
  const float* mask   = (const float*)d_in[2];
  const float* logdet = (const float*)d_in[3];
  const float* W      = (const float*)d_in[4];
  const float* bias   = (const float*)d_in[5];
  float* out      = (float*)d_out;
  float* partials = (float*)d_ws;          // 512 floats

  const size_t smem_bytes =
      (2 * A_ELEMS + 2 * B_ELEMS + ROWS_PER_BLOCK * P_STRIDE) * sizeof(float); // ~243 KB
  static bool attr_set = false;
  if (!attr_set) {
    (void)hipFuncSetAttribute((const void*)sigflow_fused_kernel,
                              hipFuncAttributeMaxDynamicSharedMemorySize,
                              (int)smem_bytes);
    attr_set = true;
  }

  sigflow_fused_kernel<<<NBLOCKS, 256, smem_bytes, stream>>>(
      x, cond, mask, W, bias, out, partials);
  sigflow_reduce_logdet<<<1, 32, 0, stream>>>(partials, logdet, out);
}